// myGAT_26182120636973
// MI455X (gfx1250) — compile-verified
//
#include <hip/hip_runtime.h>
#include <hip/hip_bf16.h>
#include <math.h>

// ---------------------------------------------------------------------------
// 3-layer single-head GATConv (PyG semantics) for MI455X / gfx1250.
//   - Dense H = Z @ W via V_WMMA_F32_16X16X4_F32 (wave32). Each wave owns a
//     16-row strip and all NT 16-col tiles: A-frag loaded once per k-step,
//     reused across NT WMMAs (4x less A traffic, 4x WMMA density per wave).
//   - Edge softmax: 3 passes with f32 atomics (max via int-order trick,
//     add via global_atomic_add_f32); scatter = one wave per edge.
// Roofline: GEMMs ~1.5 GFLOP total (negligible); edge aggregation is
// ~435MB/layer of L2-resident gather + ~54M f32 atomics/layer => the kernel is
// L2-atomic bound, fp32 end-to-end (low-precision WMMA buys nothing here).
// ---------------------------------------------------------------------------

typedef float v2f __attribute__((ext_vector_type(2)));
typedef float v8f __attribute__((ext_vector_type(8)));

#define NEG_SLOPE 0.2f

// ---------------------------------------------------------------------------
// Generic fill
// ---------------------------------------------------------------------------
__global__ __launch_bounds__(256)
void gat_fill_f32(float* __restrict__ p, float v, int n) {
    int i = blockIdx.x * blockDim.x + threadIdx.x;
    if (i < n) p[i] = v;
}

// ---------------------------------------------------------------------------
// GEMM: H[N x NT*16] = X[N x K] @ W[K x C], columns C..NT*16-1 of H set to 0.
// One wave per 16-row strip; NT accumulator tiles per wave.
// Requires N % 16 == 0 (true: N = 50000).
// WMMA f32 16x16x4 layouts (ISA 7.12.2):
//   A 16x4 : lane L holds M = L&15, K = (L>>4)*2 + v   (v = VGPR 0/1)
//   B 4x16 : lane L holds N = L&15, K = (L>>4)*2 + v   (row-striped mirror)
//   C/D    : VGPR r, lanes 0-15: (M=r, N=lane); lanes 16-31: (M=8+r, N=lane-16)
// EXEC must be all-ones: strip guard is wave-uniform, B-column masking is a
// branch-free select (clamped load * scale), so no divergence reaches the WMMA.
// ---------------------------------------------------------------------------
template <int NT>
__global__ __launch_bounds__(256)
void gat_gemm_wmma(const float* __restrict__ X, const float* __restrict__ W,
                   float* __restrict__ H, int N, int K, int C) {
    const int ldH  = NT * 16;
    const int lane = threadIdx.x & 31;
    const int wave = blockIdx.x * (blockDim.x >> 5) + (threadIdx.x >> 5);
    const int mtiles = N >> 4;
    if (wave >= mtiles) return;            // wave-uniform: EXEC stays full

    const int m0   = wave << 4;
    const int mrow = m0 + (lane & 15);     // A row for this lane
    const int kh   = (lane >> 4) << 1;     // K sub-offset (0 or 2)

    int   bcol[NT];
    float bsc[NT];
#pragma unroll
    for (int t = 0; t < NT; ++t) {
        int bn  = t * 16 + (lane & 15);    // B column for this lane, tile t
        bcol[t] = (bn < C) ? bn : (C - 1); // clamped (always in-bounds)
        bsc[t]  = (bn < C) ? 1.0f : 0.0f;  // zero pad columns >= C
    }

    v8f acc[NT] = {};
    const float* __restrict__ xrow = X + (size_t)mrow * K;
    for (int k0 = 0; k0 < K; k0 += 4) {
        const int ka = k0 + kh;
        v2f a;
        a.x = xrow[ka];
        a.y = xrow[ka + 1];
        const float* __restrict__ w0 = W + (size_t)ka * C;
        const float* __restrict__ w1 = w0 + C;
#pragma unroll
        for (int t = 0; t < NT; ++t) {
            v2f b;
            b.x = w0[bcol[t]] * bsc[t];
            b.y = w1[bcol[t]] * bsc[t];
            acc[t] = __builtin_amdgcn_wmma_f32_16x16x4_f32(
                /*neg_a=*/false, a, /*neg_b=*/false, b,
                /*c_mod=*/(short)0, acc[t], /*reuse_a=*/false, /*reuse_b=*/false);
        }
    }

    const int r0 = m0 + ((lane >> 4) << 3);
#pragma unroll
    for (int t = 0; t < NT; ++t) {
        const int cc = t * 16 + (lane & 15);
#pragma unroll
        for (int r = 0; r < 8; ++r)
            H[(size_t)(r0 + r) * ldH + cc] = acc[t][r];
    }
}

// ---------------------------------------------------------------------------
// Per-node attention scores: as[i] = dot(H[i,:C], a_src), ad[i] = dot(H[i,:C], a_dst)
// One wave per node: coalesced row read + shuffle tree reduction (wave32).
// ---------------------------------------------------------------------------
__global__ __launch_bounds__(256)
void gat_scores(const float* __restrict__ H, int ldH, int C,
                const float* __restrict__ a_src, const float* __restrict__ a_dst,
                float* __restrict__ as_, float* __restrict__ ad_, int N) {
    int gw   = (blockIdx.x * blockDim.x + threadIdx.x) >> 5;
    int lane = threadIdx.x & 31;
    if (gw >= N) return;
    const float* __restrict__ row = H + (size_t)gw * ldH;
    float s = 0.0f, d = 0.0f;
    for (int c = lane; c < C; c += 32) {
        float h = row[c];
        s += h * a_src[c];
        d += h * a_dst[c];
    }
#pragma unroll
    for (int off = 16; off > 0; off >>= 1) {
        s += __shfl_down(s, off, 32);
        d += __shfl_down(d, off, 32);
    }
    if (lane == 0) { as_[gw] = s; ad_[gw] = d; }
}

// ---------------------------------------------------------------------------
// Float atomic max via signed/unsigned int ordering trick (init must be -inf).
// ---------------------------------------------------------------------------
__device__ __forceinline__ void atomicMaxF32(float* addr, float v) {
    if (v >= 0.0f) atomicMax((int*)addr, __float_as_int(v));
    else           atomicMin((unsigned int*)addr, __float_as_uint(v));
}

__device__ __forceinline__ void edge_endpoints(const int* __restrict__ eidx,
                                               int e, int E, int& s, int& d) {
    if (e < E) { s = eidx[e]; d = eidx[E + e]; }
    else       { s = d = e - E; }              // appended self-loops
}

__device__ __forceinline__ float leaky(float v) {
    return v > 0.0f ? v : NEG_SLOPE * v;
}

// Pass 1: segment max of leaky_relu(as[src] + ad[dst]) grouped by dst
__global__ __launch_bounds__(256)
void gat_edge_max(const int* __restrict__ eidx, int E, int ET,
                  const float* __restrict__ as_, const float* __restrict__ ad_,
                  float* __restrict__ mx) {
    int e = blockIdx.x * blockDim.x + threadIdx.x;
    if (e >= ET) return;
    int s, d;
    edge_endpoints(eidx, e, E, s, d);
    atomicMaxF32(&mx[d], leaky(as_[s] + ad_[d]));
}

// Pass 2: segment sum of exp(e - max[dst])
__global__ __launch_bounds__(256)
void gat_edge_sum(const int* __restrict__ eidx, int E, int ET,
                  const float* __restrict__ as_, const float* __restrict__ ad_,
                  const float* __restrict__ mx, float* __restrict__ sm) {
    int e = blockIdx.x * blockDim.x + threadIdx.x;
    if (e >= ET) return;
    int s, d;
    edge_endpoints(eidx, e, E, s, d);
    float v = leaky(as_[s] + ad_[d]);
    atomicAdd(&sm[d], expf(v - mx[d]));
}

// Pass 3: O[dst,:] += H[src,:] * softmax-weight. One wave per edge, lanes
// stride the feature dimension (wave32 on gfx1250).
__global__ __launch_bounds__(256)
void gat_edge_scatter(const int* __restrict__ eidx, int E, int ET,
                      const float* __restrict__ as_, const float* __restrict__ ad_,
                      const float* __restrict__ mx, const float* __restrict__ sm,
                      const float* __restrict__ H, int ldH, int C,
                      float* __restrict__ O, int ldO) {
    int gw   = (blockIdx.x * blockDim.x + threadIdx.x) >> 5;
    int lane = threadIdx.x & 31;
    if (gw >= ET) return;
    int s, d;
    edge_endpoints(eidx, gw, E, s, d);
    float v = leaky(as_[s] + ad_[d]);
    float w = expf(v - mx[d]) / sm[d];
    const float* __restrict__ hrow = H + (size_t)s * ldH;
    float* __restrict__ orow = O + (size_t)d * ldO;
    for (int c = lane; c < C; c += 32)
        atomicAdd(&orow[c], hrow[c] * w);
}

// ---------------------------------------------------------------------------
// Epilogues
// ---------------------------------------------------------------------------
__global__ __launch_bounds__(256)
void gat_bias_relu(float* __restrict__ O, const float* __restrict__ b,
                   int N, int C) {
    int i = blockIdx.x * blockDim.x + threadIdx.x;
    if (i >= N * C) return;
    int c = i % C;
    float v = O[i] + b[c];
    O[i] = v > 0.0f ? v : 0.0f;
}

__global__ __launch_bounds__(256)
void gat_bias_logsoftmax(float* __restrict__ O, const float* __restrict__ b,
                         int N, int C) {
    int i = blockIdx.x * blockDim.x + threadIdx.x;
    if (i >= N) return;
    float* __restrict__ row = O + (size_t)i * C;
    float v[64];
    float m = -INFINITY;
    for (int c = 0; c < C; ++c) {
        v[c] = row[c] + b[c];
        m = fmaxf(m, v[c]);
    }
    float s = 0.0f;
    for (int c = 0; c < C; ++c) s += expf(v[c] - m);
    float ls = m + logf(s);
    for (int c = 0; c < C; ++c) row[c] = v[c] - ls;
}

// ---------------------------------------------------------------------------
// Host-side orchestration
// ---------------------------------------------------------------------------
template <int NT>
static void run_layer(const float* Z, int K, const float* W,
                      const float* a_src, const float* a_dst,
                      int C, float* Hbuf,
                      const int* eidx, int E, int ET,
                      float* as_, float* ad_, float* mx, float* sm,
                      float* O, int ldO, int N, hipStream_t stream) {
    const int B = 256;
    const int ldH = NT * 16;
    // GEMM: one wave per 16-row strip, 8 waves per block
    int strips = N >> 4;
    gat_gemm_wmma<NT><<<(strips + 7) / 8, B, 0, stream>>>(Z, W, Hbuf, N, K, C);
    // per-node scores (one wave per node)
    gat_scores<<<(N + 7) / 8, B, 0, stream>>>(Hbuf, ldH, C, a_src, a_dst,
                                              as_, ad_, N);
    // init segment buffers + output accumulator
    gat_fill_f32<<<(N + B - 1) / B, B, 0, stream>>>(mx, -INFINITY, N);
    gat_fill_f32<<<(N + B - 1) / B, B, 0, stream>>>(sm, 0.0f, N);
    gat_fill_f32<<<(N * C + B - 1) / B, B, 0, stream>>>(O, 0.0f, N * C);
    // edge softmax passes
    gat_edge_max<<<(ET + B - 1) / B, B, 0, stream>>>(eidx, E, ET, as_, ad_, mx);
    gat_edge_sum<<<(ET + B - 1) / B, B, 0, stream>>>(eidx, E, ET, as_, ad_, mx, sm);
    gat_edge_scatter<<<(ET + 7) / 8, B, 0, stream>>>(eidx, E, ET, as_, ad_, mx, sm,
                                                     Hbuf, ldH, C, O, ldO);
}

extern "C" void kernel_launch(void* const* d_in, const int* in_sizes, int n_in,
                              void* d_out, int out_size, void* d_ws, size_t ws_size,
                              hipStream_t stream) {
    const int IN = 128, HID = 64, OUT = 40, LD3 = 48;
    const int N  = in_sizes[0] / IN;       // 50000
    const int E  = in_sizes[1] / 2;        // 800000
    const int ET = E + N;                  // edges + self-loops

    const float* x     = (const float*)d_in[0];
    const int*   eidx  = (const int*)d_in[1];
    const float* W1    = (const float*)d_in[2];
    const float* asrc1 = (const float*)d_in[3];
    const float* adst1 = (const float*)d_in[4];
    const float* b1    = (const float*)d_in[5];
    const float* W2    = (const float*)d_in[6];
    const float* asrc2 = (const float*)d_in[7];
    const float* adst2 = (const float*)d_in[8];
    const float* b2    = (const float*)d_in[9];
    const float* W3    = (const float*)d_in[10];
    const float* asrc3 = (const float*)d_in[11];
    const float* adst3 = (const float*)d_in[12];
    const float* b3    = (const float*)d_in[13];
    float* out = (float*)d_out;

    // workspace layout (floats)
    float* ws  = (float*)d_ws;
    float* H1  = ws;                       // N*64
    float* O1  = H1 + (size_t)N * HID;     // N*64
    float* H2  = O1 + (size_t)N * HID;     // N*64
    float* O2  = H2 + (size_t)N * HID;     // N*64
    float* H3  = O2 + (size_t)N * HID;     // N*48 (padded, cols 40..47 are 0)
    float* as_ = H3 + (size_t)N * LD3;     // N
    float* ad_ = as_ + N;                  // N
    float* mx  = ad_ + N;                  // N
    float* sm  = mx + N;                   // N

    const int B = 256;

    // Layer 1: [N,128] -> [N,64], ReLU
    run_layer<4>(x, IN, W1, asrc1, adst1, HID, H1,
                 eidx, E, ET, as_, ad_, mx, sm, O1, HID, N, stream);
    gat_bias_relu<<<(N * HID + B - 1) / B, B, 0, stream>>>(O1, b1, N, HID);

    // Layer 2: [N,64] -> [N,64], ReLU
    run_layer<4>(O1, HID, W2, asrc2, adst2, HID, H2,
                 eidx, E, ET, as_, ad_, mx, sm, O2, HID, N, stream);
    gat_bias_relu<<<(N * HID + B - 1) / B, B, 0, stream>>>(O2, b2, N, HID);

    // Layer 3: [N,64] -> [N,40] (H padded to ld=48), log_softmax into d_out
    run_layer<3>(O2, HID, W3, asrc3, adst3, OUT, H3,
                 eidx, E, ET, as_, ad_, mx, sm, out, OUT, N, stream);
    gat_bias_logsoftmax<<<(N + B - 1) / B, B, 0, stream>>>(out, b3, N, OUT);
}